// GridSamplePScan_7069516169775
// MI455X (gfx1250) — compile-verified
//
#include <hip/hip_runtime.h>

#ifndef __has_builtin
#define __has_builtin(x) 0
#endif

// Problem shape (fixed by the reference)
#define B_ 4
#define L_ 32
#define C_ 16
#define H_ 64
#define W_ 64
#define HW_ (H_ * W_)           // 4096
// Tiling
#define CPB 4                   // channels per block (C_/CPB blocks per (b,t))
#define NBLK_C (C_ / CPB)       // 4
#define THREADS_ 1024           // 32 wave32 waves per workgroup
#define NPIX (HW_ / THREADS_)   // 4 pixels per thread
#define LDS_FLOATS (CPB * HW_)  // 16384 floats = 64 KB per buffer
#define STAGE_ROUNDS (LDS_FLOATS / (THREADS_ * 4))  // 4 x b128 per thread

#if defined(__AMDGCN__) && __has_builtin(__builtin_amdgcn_global_load_async_to_lds_b128)
#define HAVE_ASYNC_LDS 1
#else
#define HAVE_ASYNC_LDS 0
#endif

typedef int v4i_ __attribute__((ext_vector_type(4)));

// ---------------------------------------------------------------------------
// Kernel 1: cum = cumsum(flows, axis=L). flows: [B, L, 2, H, W] fp32.
// One thread per (b, comp, pixel); serial scan over L (32 steps).
// ---------------------------------------------------------------------------
__global__ void gsps_cumsum_kernel(const float* __restrict__ flows,
                                   float* __restrict__ cum) {
  int g = blockIdx.x * blockDim.x + threadIdx.x;   // [0, B*2*HW)
  if (g >= B_ * 2 * HW_) return;
  int pix  = g % HW_;
  int comp = (g / HW_) & 1;
  int b    = g / (2 * HW_);
  size_t base   = (((size_t)b * L_) * 2 + comp) * HW_ + pix;
  size_t stride = (size_t)2 * HW_;
  float acc = 0.0f;
#pragma unroll
  for (int l = 0; l < L_; ++l) {
    acc += flows[base + (size_t)l * stride];
    cum[base + (size_t)l * stride] = acc;
  }
}

// ---------------------------------------------------------------------------
// Kernel 2: for block (b, t, q): out[b,t, q*CPB..q*CPB+3, :, :] =
//   sum_{k<=t} bilinear(images[b,k,c], wrap(base + cum[t]-cum[k]))
// Double-buffered async global->LDS pipeline: while gathering from buffer
// k&1, the k+1 slab streams into buffer (k+1)&1 via
// global_load_async_to_lds_b128 (ASYNCcnt), hidden under the per-pixel
// coordinate math + LDS gathers. Slab k+2 is software-prefetched into L2.
// ---------------------------------------------------------------------------
__global__ __launch_bounds__(THREADS_) void gsps_warp_scan_kernel(
    const float* __restrict__ images,  // [B, L, C, H, W]
    const float* __restrict__ cum,     // [B, L, 2, H, W]
    float* __restrict__ out) {         // [B, L, C, H, W]
  __shared__ float smem[2 * LDS_FLOATS];  // 2 x 64 KB double buffer

  const int tid = threadIdx.x;
  const int q = blockIdx.x % NBLK_C;
  const int t = (blockIdx.x / NBLK_C) % L_;
  const int b = blockIdx.x / (NBLK_C * L_);
  const int c0 = q * CPB;

  const float* slab0 = images + (((size_t)b * L_) * C_ + c0) * HW_;
  const size_t kstride = (size_t)C_ * HW_;  // advance k by 1

  // Per-pixel: a = base_coord + cum_t (coordinate before subtracting cum_k)
  float ax[NPIX], ay[NPIX];
  float acc[NPIX][CPB];
#pragma unroll
  for (int p = 0; p < NPIX; ++p) {
    int pix = p * THREADS_ + tid;          // adjacent lanes -> adjacent w
    int w = pix & (W_ - 1);
    int h = pix >> 6;
    float bx = ((float)w + 0.5f) * (2.0f / W_) - 1.0f;
    float by = ((float)h + 0.5f) * (2.0f / H_) - 1.0f;
    size_t ct = (((size_t)b * L_ + t) * 2) * HW_ + pix;
    ax[p] = bx + cum[ct];
    ay[p] = by + cum[ct + HW_];
#pragma unroll
    for (int c = 0; c < CPB; ++c) acc[p][c] = 0.0f;
  }

  // ---- Stage one 64 KB slab into LDS buffer `buf` (16 B/lane/issue) ----
  auto stage = [&](int buf, int k) {
    const float* src = slab0 + (size_t)k * kstride;
    float* dst = &smem[buf * LDS_FLOATS];
#if HAVE_ASYNC_LDS
#pragma unroll
    for (int r = 0; r < STAGE_ROUNDS; ++r) {
      int idx4 = (r * THREADS_ + tid) * 4;
      __builtin_amdgcn_global_load_async_to_lds_b128(
          (__attribute__((address_space(1))) v4i_*)(src + idx4),
          (__attribute__((address_space(3))) v4i_*)(dst + idx4),
          0, 0);
    }
#else
#pragma unroll
    for (int r = 0; r < STAGE_ROUNDS; ++r) {
      int idx4 = (r * THREADS_ + tid) * 4;
      *(float4*)(dst + idx4) = *(const float4*)(src + idx4);
    }
#endif
  };
  auto wait_async = [&]() {
#if HAVE_ASYNC_LDS
#if __has_builtin(__builtin_amdgcn_s_wait_asynccnt)
    __builtin_amdgcn_s_wait_asynccnt(0);
#else
    asm volatile("s_wait_asynccnt 0" ::: "memory");
#endif
#endif
  };

  // Prologue: fill buffer 0 with slab k=0.
  stage(0, 0);
  wait_async();
  __syncthreads();

  for (int k = 0; k <= t; ++k) {
    // Start streaming the next slab into the other buffer (overlapped).
    if (k < t) stage((k + 1) & 1, k + 1);

    // Prefetch slab k+2 into cache (512 lines of 128 B).
    if (k + 2 <= t && tid < (LDS_FLOATS * 4) / 128) {
      const char* nsrc = (const char*)(slab0 + (size_t)(k + 2) * kstride);
      __builtin_prefetch(nsrc + tid * 128, 0, 0);
    }

    // ---- Gather + accumulate from buffer k&1 ----
    const float* sbuf = &smem[(k & 1) * LDS_FLOATS];
    size_t ck = (((size_t)b * L_ + k) * 2) * HW_;
#pragma unroll
    for (int p = 0; p < NPIX; ++p) {
      int pix = p * THREADS_ + tid;
      float relx = ax[p] - cum[ck + pix];
      float rely = ay[p] - cum[ck + HW_ + pix];
      // wrap to [-1, 1):  mod(v+1, 2) - 1  with non-negative mod
      float txv = relx + 1.0f;
      float gx = (txv - 2.0f * floorf(txv * 0.5f)) - 1.0f;
      float tyv = rely + 1.0f;
      float gy = (tyv - 2.0f * floorf(tyv * 0.5f)) - 1.0f;
      float ixf = (gx + 1.0f) * 0.5f * (float)W_ - 0.5f;
      float iyf = (gy + 1.0f) * 0.5f * (float)H_ - 0.5f;
      float fx0 = floorf(ixf), fy0 = floorf(iyf);
      int x0 = (int)fx0, y0 = (int)fy0;
      int x1 = x0 + 1, y1 = y0 + 1;
      float wx1 = ixf - fx0, wy1 = iyf - fy0;
      float wx0 = 1.0f - wx1, wy0 = 1.0f - wy1;
      // validity (zero padding outside image); fold into the weights
      float vx0 = (x0 >= 0 && x0 < W_) ? 1.0f : 0.0f;
      float vx1 = (x1 >= 0 && x1 < W_) ? 1.0f : 0.0f;
      float vy0 = (y0 >= 0 && y0 < H_) ? 1.0f : 0.0f;
      float vy1 = (y1 >= 0 && y1 < H_) ? 1.0f : 0.0f;
      float w00 = wx0 * wy0 * vx0 * vy0;
      float w10 = wx1 * wy0 * vx1 * vy0;
      float w01 = wx0 * wy1 * vx0 * vy1;
      float w11 = wx1 * wy1 * vx1 * vy1;
      int x0c = min(max(x0, 0), W_ - 1), x1c = min(max(x1, 0), W_ - 1);
      int y0c = min(max(y0, 0), H_ - 1), y1c = min(max(y1, 0), H_ - 1);
      int o00 = y0c * W_ + x0c;
      int o10 = y0c * W_ + x1c;
      int o01 = y1c * W_ + x0c;
      int o11 = y1c * W_ + x1c;
#pragma unroll
      for (int c = 0; c < CPB; ++c) {
        const float* sp = sbuf + c * HW_;
        acc[p][c] += w00 * sp[o00] + w10 * sp[o10] +
                     w01 * sp[o01] + w11 * sp[o11];
      }
    }

    // k+1 slab copies finished + all gathers of this slab done.
    wait_async();
    __syncthreads();
  }

  // ---- Write result ----
#pragma unroll
  for (int p = 0; p < NPIX; ++p) {
    int pix = p * THREADS_ + tid;
    size_t ob = (((size_t)b * L_ + t) * C_ + c0) * HW_ + pix;
#pragma unroll
    for (int c = 0; c < CPB; ++c) {
      out[ob + (size_t)c * HW_] = acc[p][c];
    }
  }
}

// ---------------------------------------------------------------------------
extern "C" void kernel_launch(void* const* d_in, const int* in_sizes, int n_in,
                              void* d_out, int out_size, void* d_ws, size_t ws_size,
                              hipStream_t stream) {
  const float* flows  = (const float*)d_in[0];  // [B, L, 2, H, W] fp32
  const float* images = (const float*)d_in[1];  // [B, L, C, H, W] fp32
  float* cum = (float*)d_ws;                    // [B, L, 2, H, W] fp32 (4 MB)
  float* out = (float*)d_out;                   // [B, L, C, H, W] fp32

  // Kernel 1: cumulative flow
  {
    int n = B_ * 2 * HW_;  // 32768 threads
    gsps_cumsum_kernel<<<(n + 255) / 256, 256, 0, stream>>>(flows, cum);
  }
  // Kernel 2: warp + masked scan accumulate
  {
    dim3 grid(B_ * L_ * NBLK_C);  // 512 blocks
    gsps_warp_scan_kernel<<<grid, THREADS_, 0, stream>>>(images, cum, out);
  }
}